// ScaledDotProductAttention_90297392431113
// MI455X (gfx1250) — compile-verified
//
#include <hip/hip_runtime.h>

typedef __attribute__((ext_vector_type(2))) float v2f;
typedef __attribute__((ext_vector_type(8))) float v8f;

#define BH   64
#define SEQ  2048
#define DIM  128
#define TEMP_INV 0.08838834764831845f   // 1/sqrt(128)
#define NEG_INF  -1.0e9f

// LDS row strides (floats), padded for bank-conflict-free fragment loads
#define QS 132
#define KS 132
#define SS 2052
#define VS 132
#define WAVES 8

// shared-memory float offsets
#define Q_OFF  0
#define SC_OFF (16 * QS)                       // 2112
#define KV_OFF (SC_OFF + 16 * SS)              // 34944
#define KV_PER_WAVE (2 * 16 * KS)              // double-buffered K slab per wave
#define SMEM_FLOATS (KV_OFF + WAVES * KV_PER_WAVE) // 68736 floats = 274,944 B < 320 KB

// CDNA5 async global->LDS copy (ASYNCcnt-tracked, no VGPR round trip).
// vdst = 32-bit LDS byte offset, vaddr = 64-bit global address.
__device__ __forceinline__ void async_ld_b128(const float* g, float* l) {
    unsigned lds = (unsigned)(size_t)l;   // low 32 bits of AS3->flat cast = LDS offset
    asm volatile("global_load_async_to_lds_b128 %0, %1, off"
                 :: "v"(lds), "v"(g) : "memory");
}
#define WAIT_ASYNC_0()  asm volatile("s_wait_asynccnt 0"  ::: "memory")
#define WAIT_ASYNC_16() asm volatile("s_wait_asynccnt 16" ::: "memory")

__global__ __launch_bounds__(256, 1)
void sdpa_wmma_f32_kernel(const float* __restrict__ q,
                          const float* __restrict__ k,
                          const float* __restrict__ v,
                          const unsigned char* __restrict__ mask,
                          float* __restrict__ out,
                          float* __restrict__ attn)
{
    extern __shared__ float smem[];
    float* q_s  = smem + Q_OFF;   // 16 x 128 (stride QS)
    float* sc_s = smem + SC_OFF;  // 16 x 2048 (stride SS)
    float* kv_s = smem + KV_OFF;  // per-wave double K slabs / shared double V chunks

    const int tid  = threadIdx.x;
    const int wave = tid >> 5;
    const int lane = tid & 31;
    const int bh   = blockIdx.y;
    const int qr0  = blockIdx.x * 16;

    const int m    = lane & 15;        // fragment row (A) / col (B)
    const int koff = (lane >> 4) << 1; // 0 or 2 : K-subindex per ISA 16x4 layout

    float* kb0 = kv_s + wave * KV_PER_WAVE;
    float* kb1 = kb0 + 16 * KS;

    // ---- async-stage Q tile (2 b128/thread), then K tile 0 (16 b128/lane) ----
    {
        const float* qg = q + ((size_t)bh * SEQ + qr0) * DIM;
        #pragma unroll
        for (int i = 0; i < 2; ++i) {
            int idx = tid + i * 256;  // float4 index 0..511
            int row = idx >> 5, c4 = idx & 31;
            async_ld_b128(qg + row * DIM + c4 * 4, q_s + row * QS + c4 * 4);
        }
        const float* kg = k + ((size_t)bh * SEQ + (wave * 16) * 16) * DIM;
        #pragma unroll
        for (int r = 0; r < 16; ++r)
            async_ld_b128(kg + r * DIM + lane * 4, kb0 + r * KS + lane * 4);
    }
    WAIT_ASYNC_16();   // in-order: drains the 2 Q loads (18 outstanding -> 16)
    __syncthreads();   // Q tile visible block-wide

    // ---- Phase 1: scores = mask(QK^T / T); double-buffered K, no barriers ----
    for (int t = 0; t < 16; ++t) {
        float* cur = (t & 1) ? kb1 : kb0;
        if (t < 15) {   // prefetch next K tile into the other slab
            float* nxt = (t & 1) ? kb0 : kb1;
            const float* kg = k + ((size_t)bh * SEQ + (wave * 16 + t + 1) * 16) * DIM;
            #pragma unroll
            for (int r = 0; r < 16; ++r)
                async_ld_b128(kg + r * DIM + lane * 4, nxt + r * KS + lane * 4);
            WAIT_ASYNC_16();   // previous tile's 16 loads have landed
        } else {
            WAIT_ASYNC_0();
        }
        v8f c = {};
        #pragma unroll
        for (int kk = 0; kk < DIM; kk += 4) {
            v2f a = *(const v2f*)(q_s + m * QS + kk + koff);  // Q[m][kk+koff..+1]
            v2f b = *(const v2f*)(cur + m * KS + kk + koff);  // K[n0+m][..] == B[k][n]
            c = __builtin_amdgcn_wmma_f32_16x16x4_f32(false, a, false, b,
                                                      (short)0, c, false, false);
        }
        const int n0  = (wave * 16 + t) * 16;
        const int col = n0 + m;
        const unsigned char* mrow = mask + ((size_t)bh * SEQ + qr0) * SEQ + col;
        #pragma unroll
        for (int r = 0; r < 8; ++r) {
            int row = r + ((lane >> 4) << 3);
            float s = c[r] * TEMP_INV;
            sc_s[row * SS + col] = mrow[(size_t)row * SEQ] ? NEG_INF : s;
        }
    }
    __syncthreads();

    // ---- prefetch V chunk 0 now so DMA overlaps the softmax phase ----
    float* vb0 = kv_s;
    float* vb1 = kv_s + 32 * VS;
    {
        const float* vg = v + (size_t)bh * SEQ * DIM;
        #pragma unroll
        for (int i = 0; i < 4; ++i) {
            int idx = tid + i * 256;  // float4 index 0..1023 (32 rows x 32 f4)
            int row = idx >> 5, c4 = idx & 31;
            async_ld_b128(vg + row * DIM + c4 * 4, vb0 + row * VS + c4 * 4);
        }
    }

    // ---- Phase 2: softmax, 2 rows per wave; write attn out coalesced ----
    #pragma unroll
    for (int rr = 0; rr < 2; ++rr) {
        const int row = wave * 2 + rr;
        float* sr = sc_s + row * SS;
        float mx = -3.402823466e38f;
        for (int j = lane; j < SEQ; j += 32) mx = fmaxf(mx, sr[j]);
        #pragma unroll
        for (int off = 16; off; off >>= 1) mx = fmaxf(mx, __shfl_xor(mx, off, 32));
        float sum = 0.f;
        for (int j = lane; j < SEQ; j += 32) {
            float e = __expf(sr[j] - mx);
            sr[j] = e;
            sum += e;
        }
        #pragma unroll
        for (int off = 16; off; off >>= 1) sum += __shfl_xor(sum, off, 32);
        const float inv = 1.0f / sum;
        float* ag = attn + ((size_t)bh * SEQ + qr0 + row) * SEQ;
        for (int j = lane; j < SEQ; j += 32) {
            float p = sr[j] * inv;
            sr[j] = p;
            ag[j] = p;
        }
    }

    // ---- Phase 3: out = P @ V ; double-buffered chunks, 1 barrier/chunk ----
    const int n0 = wave * 16;
    v8f acc = {};
    for (int ck = 0; ck < SEQ / 32; ++ck) {
        WAIT_ASYNC_0();       // my chunk-ck loads have landed
        __syncthreads();      // everyone's landed; prev buffer fully consumed
        if (ck < SEQ / 32 - 1) {
            float* nxt = ((ck + 1) & 1) ? vb1 : vb0;
            const float* vg = v + ((size_t)bh * SEQ + (ck + 1) * 32) * DIM;
            #pragma unroll
            for (int i = 0; i < 4; ++i) {
                int idx = tid + i * 256;
                int row = idx >> 5, c4 = idx & 31;
                async_ld_b128(vg + row * DIM + c4 * 4, nxt + row * VS + c4 * 4);
            }
        }
        float* cur = (ck & 1) ? vb1 : vb0;
        #pragma unroll
        for (int kk = 0; kk < 32; kk += 4) {
            v2f a = *(const v2f*)(sc_s + m * SS + ck * 32 + kk + koff); // P fragment
            v2f b;
            b[0] = cur[(kk + koff)     * VS + n0 + m];                  // V[k][n]
            b[1] = cur[(kk + koff + 1) * VS + n0 + m];
            acc = __builtin_amdgcn_wmma_f32_16x16x4_f32(false, a, false, b,
                                                        (short)0, acc, false, false);
        }
    }
    #pragma unroll
    for (int r = 0; r < 8; ++r) {
        int row = r + ((lane >> 4) << 3);
        out[((size_t)bh * SEQ + qr0 + row) * DIM + n0 + m] = acc[r];
    }
}

extern "C" void kernel_launch(void* const* d_in, const int* in_sizes, int n_in,
                              void* d_out, int out_size, void* d_ws, size_t ws_size,
                              hipStream_t stream) {
    (void)in_sizes; (void)n_in; (void)out_size; (void)d_ws; (void)ws_size;
    const float* q = (const float*)d_in[0];
    const float* k = (const float*)d_in[1];
    const float* v = (const float*)d_in[2];
    const unsigned char* mask = (const unsigned char*)d_in[3];  // jax bool -> 1 byte
    float* out  = (float*)d_out;
    float* attn = out + (size_t)BH * SEQ * DIM;   // tuple: (output, attn) flat-concat

    dim3 grid(SEQ / 16, BH);
    size_t smem_bytes = (size_t)SMEM_FLOATS * sizeof(float);
    sdpa_wmma_f32_kernel<<<grid, 256, smem_bytes, stream>>>(q, k, v, mask, out, attn);
}